// DownSampleBlock_8976481648903
// MI455X (gfx1250) — compile-verified
//
#include <hip/hip_runtime.h>
#include <hip/hip_bf16.h>

// ---------------------------------------------------------------------------
// DownSampleBlock for MI455X (gfx1250, wave32, WMMA + TDM).
//   Stage A (mlp_kernel): fused  f@w1 -> LN -> relu -> @w2 (+b2) -> h
//                                 h@dw1 -> LN -> relu -> @dw2 -> relu -> w
//     - 8 waves/block, one 16-row tile per wave, 256 blocks.
//     - w1 / dw1 staged to LDS by the Tensor Data Mover (tensor_load_to_lds)
//       with hardware row padding (pad_enable) -> conflict-free +1 rows.
//     - All GEMMs on V_WMMA_F32_16X16X4_F32; B-fragments preloaded into
//       register arrays so WMMAs issue back-to-back instead of
//       load->wait->wmma chains.
//     - LN via 16-lane shfl_xor butterflies in the WMMA C-layout.
//   Stage B (topk_gather_kernel): continuous_topk forward == stable sort
//     (value desc, index asc); bitonic sort of 2048 keys per batch in LDS,
//     then gather xyz / h / indices.
// ---------------------------------------------------------------------------

typedef __attribute__((ext_vector_type(2))) float v2f;
typedef __attribute__((ext_vector_type(8))) float v8f;
typedef __attribute__((ext_vector_type(4))) unsigned int v4u;
typedef __attribute__((ext_vector_type(4))) int v4i;
typedef __attribute__((ext_vector_type(8))) int v8i;

#define BATCH 16
#define NPTS  2048
#define CIN   64
#define COUT  128
#define DMID  64
#define OUT_K 1024
#define LN_EPS 1e-05f

__device__ __forceinline__ v8f wmma4(v2f a, v2f b, v8f c) {
    // D = A(16x4 f32) x B(4x16 f32) + C(16x16 f32)
    return __builtin_amdgcn_wmma_f32_16x16x4_f32(
        /*neg_a=*/false, a, /*neg_b=*/false, b,
        /*c_mod=*/(short)0, c, /*reuse_a=*/false, /*reuse_b=*/false);
}

__device__ __forceinline__ float halfsum16(float s) {
    // Sum across the 16 lanes of this wave-half (rows live per half).
    s += __shfl_xor(s, 1);
    s += __shfl_xor(s, 2);
    s += __shfl_xor(s, 4);
    s += __shfl_xor(s, 8);
    return s;
}

// TDM: 2D tile load (tile == whole tensor), 4-byte elements, with LDS row
// padding of 1 dword after every 2^(pad_code+1) dwords. D# layout per CDNA5
// ISA 8.3/8.4 (group0: count/lds/global/type, group1: dims/strides/pad).
__device__ __forceinline__ void tdm_load_2d_padded(const void* gaddr,
                                                   unsigned lds_off,
                                                   unsigned dim0, unsigned dim1,
                                                   unsigned stride0,
                                                   unsigned pad_code) {
    unsigned long long ga = (unsigned long long)gaddr;
    v4u g0;
    g0[0] = 1u;                                    // count=1, user descriptor
    g0[1] = lds_off;                               // lds_addr (bytes)
    g0[2] = (unsigned)(ga & 0xFFFFFFFFu);          // global_addr[31:0]
    g0[3] = (unsigned)((ga >> 32) & 0x01FFFFFFu)   // global_addr[56:32]
            | 0x80000000u;                         // type = 2 ("image")
    v8i g1;
    g1[0] = (int)((2u << 16)                       // data_size = 4B
                  | (1u << 20)                     // pad_enable
                  | (pad_code << 22));             // pad_interval; amount=1dw
    g1[1] = (int)(dim0 << 16);                     // tensor_dim0[15:0]
    g1[2] = (int)((dim0 >> 16) | (dim1 << 16));    // dim0[31:16] | dim1[15:0]
    g1[3] = (int)((dim1 >> 16) | (dim0 << 16));    // dim1[31:16] | tile_dim0
    g1[4] = (int)(dim1 & 0xFFFFu);                 // tile_dim1 | tile_dim2=0
    g1[5] = (int)stride0;                          // tensor_dim0_stride[31:0]
    g1[6] = 0;                                     // stride hi | dim1_stride
    g1[7] = 0;
    v4i z4 = {0, 0, 0, 0};
    v8i z8 = {0, 0, 0, 0, 0, 0, 0, 0};
    // 6-arg toolchain form: (g0, g1, g2, g3, g4, cpol)
    __builtin_amdgcn_tensor_load_to_lds(g0, g1, z4, z4, z8, /*cpol=*/0);
}

__global__ __launch_bounds__(256) void mlp_kernel(
    const float* __restrict__ features,  // (B, CIN, N)
    const float* __restrict__ w1, const float* __restrict__ b1,
    const float* __restrict__ g1, const float* __restrict__ be1,
    const float* __restrict__ w2, const float* __restrict__ b2,
    const float* __restrict__ dw1, const float* __restrict__ db1,
    const float* __restrict__ dg,  const float* __restrict__ dbe,
    const float* __restrict__ dw2, const float* __restrict__ db2,
    float* __restrict__ h_ws,   // (B, N, COUT)
    float* __restrict__ w_ws)   // (B, N)
{
    const int lane = threadIdx.x & 31;
    const int wave = threadIdx.x >> 5;
    const int tile = blockIdx.x * 8 + wave;      // 2048 tiles total
    const int b    = tile >> 7;                  // tile / (N/16)
    const int n0   = (tile & 127) << 4;
    const int l    = lane & 15;                  // column lane within half
    const int half = lane >> 4;

    // +1-padded rows -> conflict-free column reads (stride mod 64 != 0).
    __shared__ float w1s[CIN][COUT + 1];         // 33 KB, TDM-padded
    __shared__ float dw1s[COUT][DMID + 1];       // 33 KB, TDM-padded
    __shared__ float stg[8][16][COUT + 1];       // 66 KB C->A relayout tiles
    float (*hrow)[COUT + 1] = stg[wave];

    // ---- Stage weights into LDS via the Tensor Data Mover (one wave) ----
    if (threadIdx.x == 0) {
        tdm_load_2d_padded(w1, (unsigned)(unsigned long long)&w1s[0][0],
                           COUT, CIN, COUT, /*128 dwords*/ 6);
        tdm_load_2d_padded(dw1, (unsigned)(unsigned long long)&dw1s[0][0],
                           DMID, COUT, DMID, /*64 dwords*/ 5);
        __builtin_amdgcn_s_wait_tensorcnt(0);
    }
    __syncthreads();

    // ---------------- Stage 1: h1 = relu(LN(f @ w1 + b1)) ----------------
    // A-frag layout (ISA 7.12.2, 32-bit A 16x4): lanes 0-15 M=0..15,
    //   VGPR0 holds K = 4k + 2*half, VGPR1 holds K = 4k + 1 + 2*half.
    v2f af1[CIN / 4];
    {
        const float* fb = features + (size_t)b * CIN * NPTS + (n0 + l);
        #pragma unroll
        for (int k = 0; k < CIN / 4; ++k) {
            const int c0 = 4 * k + 2 * half;
            af1[k].x = fb[(size_t)c0 * NPTS];
            af1[k].y = fb[(size_t)(c0 + 1) * NPTS];
        }
    }

    v8f acc[COUT / 16];
    #pragma unroll
    for (int j = 0; j < COUT / 16; ++j) {
        v8f c = {};
        v2f bf[CIN / 4];
        #pragma unroll
        for (int k = 0; k < CIN / 4; ++k) {       // batch the LDS reads
            const int c0 = 4 * k + 2 * half;
            bf[k].x = w1s[c0][16 * j + l];
            bf[k].y = w1s[c0 + 1][16 * j + l];
        }
        #pragma unroll
        for (int k = 0; k < CIN / 4; ++k)          // back-to-back WMMAs
            c = wmma4(af1[k], bf[k], c);
        acc[j] = c;
    }

    // bias + LayerNorm(128) + relu, all in the C-layout
    {
        float b1v[8], g1v[8], bev[8];
        #pragma unroll
        for (int j = 0; j < 8; ++j) {
            b1v[j] = b1[16 * j + l];
            g1v[j] = g1[16 * j + l];
            bev[j] = be1[16 * j + l];
        }
        #pragma unroll
        for (int j = 0; j < 8; ++j)
            #pragma unroll
            for (int r = 0; r < 8; ++r) acc[j][r] += b1v[j];

        #pragma unroll
        for (int r = 0; r < 8; ++r) {
            float s = 0.f;
            #pragma unroll
            for (int j = 0; j < 8; ++j) s += acc[j][r];
            const float mean = halfsum16(s) * (1.f / 128.f);
            float vs = 0.f;
            #pragma unroll
            for (int j = 0; j < 8; ++j) {
                const float d0 = acc[j][r] - mean;
                vs += d0 * d0;
            }
            const float rstd = rsqrtf(halfsum16(vs) * (1.f / 128.f) + LN_EPS);
            #pragma unroll
            for (int j = 0; j < 8; ++j) {
                const float v = (acc[j][r] - mean) * rstd * g1v[j] + bev[j];
                acc[j][r] = fmaxf(v, 0.f);
            }
        }
    }

    // C-layout -> LDS so stage 2 can read A-frags (transposed striping).
    __syncthreads();
    #pragma unroll
    for (int j = 0; j < 8; ++j)
        #pragma unroll
        for (int r = 0; r < 8; ++r)
            hrow[r + 8 * half][16 * j + l] = acc[j][r];
    __syncthreads();

    // ---------------- Stage 2: h = h1 @ w2 + b2 ----------------
    v2f af2[COUT / 4];
    #pragma unroll
    for (int k = 0; k < COUT / 4; ++k) {
        const int c0 = 4 * k + 2 * half;
        af2[k].x = hrow[l][c0];
        af2[k].y = hrow[l][c0 + 1];
    }
    v8f acc2[COUT / 16];
    #pragma unroll
    for (int j = 0; j < COUT / 16; ++j) {
        v8f c = {};
        #pragma unroll
        for (int ch = 0; ch < 4; ++ch) {           // 8-deep register chunks
            v2f bf[8];
            #pragma unroll
            for (int kk = 0; kk < 8; ++kk) {       // batch the global reads
                const int c0 = 4 * (8 * ch + kk) + 2 * half;
                const float* wb = w2 + (size_t)c0 * COUT + 16 * j + l;
                bf[kk].x = wb[0];
                bf[kk].y = wb[COUT];
            }
            #pragma unroll
            for (int kk = 0; kk < 8; ++kk)
                c = wmma4(af2[8 * ch + kk], bf[kk], c);
        }
        acc2[j] = c;
    }
    #pragma unroll
    for (int j = 0; j < 8; ++j) {
        const float b2v = b2[16 * j + l];
        #pragma unroll
        for (int r = 0; r < 8; ++r) acc2[j][r] += b2v;
    }

    // Store h to HBM (gathered later) and to LDS for stage 3.
    {
        float* hb = h_ws + ((size_t)b * NPTS + n0) * COUT;
        #pragma unroll
        for (int j = 0; j < 8; ++j)
            #pragma unroll
            for (int r = 0; r < 8; ++r)
                hb[(size_t)(r + 8 * half) * COUT + 16 * j + l] = acc2[j][r];
    }
    __syncthreads();
    #pragma unroll
    for (int j = 0; j < 8; ++j)
        #pragma unroll
        for (int r = 0; r < 8; ++r)
            hrow[r + 8 * half][16 * j + l] = acc2[j][r];
    __syncthreads();

    // ---------------- Stage 3: d = relu(LN(h @ dw1 + db1)) ----------------
    v2f af3[COUT / 4];
    #pragma unroll
    for (int k = 0; k < COUT / 4; ++k) {
        const int c0 = 4 * k + 2 * half;
        af3[k].x = hrow[l][c0];
        af3[k].y = hrow[l][c0 + 1];
    }
    v8f acc3[DMID / 16];
    #pragma unroll
    for (int j = 0; j < DMID / 16; ++j) {
        v8f c = {};
        #pragma unroll
        for (int ch = 0; ch < 2; ++ch) {           // 16-deep register chunks
            v2f bf[16];
            #pragma unroll
            for (int kk = 0; kk < 16; ++kk) {      // batch the LDS reads
                const int c0 = 4 * (16 * ch + kk) + 2 * half;
                bf[kk].x = dw1s[c0][16 * j + l];
                bf[kk].y = dw1s[c0 + 1][16 * j + l];
            }
            #pragma unroll
            for (int kk = 0; kk < 16; ++kk)
                c = wmma4(af3[16 * ch + kk], bf[kk], c);
        }
        acc3[j] = c;
    }
    {
        float dbv[4], dgv[4], dbev[4];
        #pragma unroll
        for (int j = 0; j < 4; ++j) {
            dbv[j]  = db1[16 * j + l];
            dgv[j]  = dg[16 * j + l];
            dbev[j] = dbe[16 * j + l];
        }
        #pragma unroll
        for (int j = 0; j < 4; ++j)
            #pragma unroll
            for (int r = 0; r < 8; ++r) acc3[j][r] += dbv[j];

        #pragma unroll
        for (int r = 0; r < 8; ++r) {
            float s = 0.f;
            #pragma unroll
            for (int j = 0; j < 4; ++j) s += acc3[j][r];
            const float mean = halfsum16(s) * (1.f / 64.f);
            float vs = 0.f;
            #pragma unroll
            for (int j = 0; j < 4; ++j) {
                const float d0 = acc3[j][r] - mean;
                vs += d0 * d0;
            }
            const float rstd = rsqrtf(halfsum16(vs) * (1.f / 64.f) + LN_EPS);
            #pragma unroll
            for (int j = 0; j < 4; ++j) {
                const float v = (acc3[j][r] - mean) * rstd * dgv[j] + dbev[j];
                acc3[j][r] = fmaxf(v, 0.f);
            }
        }
    }

    // ---------------- Stage 4: w = relu(d @ dw2 + db2) ----------------
    {
        float dw2v[4];
        #pragma unroll
        for (int j = 0; j < 4; ++j) dw2v[j] = dw2[16 * j + l];
        const float db2v = db2[0];
        #pragma unroll
        for (int r = 0; r < 8; ++r) {
            float t = 0.f;
            #pragma unroll
            for (int j = 0; j < 4; ++j) t += acc3[j][r] * dw2v[j];
            t = halfsum16(t);
            const float wv = fmaxf(t + db2v, 0.f);
            if (l == r)
                w_ws[(size_t)b * NPTS + n0 + r + 8 * half] = wv;
        }
    }
}

// ---------------------------------------------------------------------------
// Top-k (forward of continuous_topk) == stable sort (value desc, index asc),
// take first OUT_K. One 1024-thread block per batch, bitonic sort in LDS,
// then gather xyz / h / indices.
// ---------------------------------------------------------------------------
__global__ __launch_bounds__(1024) void topk_gather_kernel(
    const float* __restrict__ w_ws,   // (B, N)
    const float* __restrict__ h_ws,   // (B, N, COUT)
    const float* __restrict__ xyzs,   // (B, N, 3)
    float* __restrict__ xyz_out,      // (B, OUT_K, 3)
    float* __restrict__ feats_out,    // (B, COUT, OUT_K)
    int*   __restrict__ idx_out)      // (B, OUT_K)
{
    const int b = blockIdx.x;
    const int t = threadIdx.x;

    __shared__ float sv[NPTS];
    __shared__ int   si[NPTS];

    sv[t]          = w_ws[(size_t)b * NPTS + t];
    si[t]          = t;
    sv[t + OUT_K]  = w_ws[(size_t)b * NPTS + t + OUT_K];
    si[t + OUT_K]  = t + OUT_K;
    __syncthreads();

    // Bitonic sort, comparator: a before b  <=>  a.v > b.v || (== && a.i < b.i)
    for (int k = 2; k <= NPTS; k <<= 1) {
        for (int j = k >> 1; j > 0; j >>= 1) {
            const int i = 2 * t - (t & (j - 1));
            const int p = i + j;
            const bool up = ((i & k) == 0);
            const float av = sv[i], bv = sv[p];
            const int   ai = si[i], bi = si[p];
            const bool before = (av > bv) || (av == bv && ai < bi);
            if (before != up) {
                sv[i] = bv; sv[p] = av;
                si[i] = bi; si[p] = ai;
            }
            __syncthreads();
        }
    }

    // Emit outputs (first OUT_K sorted entries == selection order).
    const int sel = si[t];
    idx_out[(size_t)b * OUT_K + t] = sel;

    {
        const float* xs = xyzs + ((size_t)b * NPTS + sel) * 3;
        float* xo = xyz_out + ((size_t)b * OUT_K + t) * 3;
        xo[0] = xs[0]; xo[1] = xs[1]; xo[2] = xs[2];
    }
    {
        const float* hb = h_ws + ((size_t)b * NPTS + sel) * COUT;
        float* fo = feats_out + (size_t)b * COUT * OUT_K + t;
        #pragma unroll 4
        for (int c = 0; c < COUT; ++c)
            fo[(size_t)c * OUT_K] = hb[c];
    }
}

extern "C" void kernel_launch(void* const* d_in, const int* in_sizes, int n_in,
                              void* d_out, int out_size, void* d_ws, size_t ws_size,
                              hipStream_t stream) {
    const float* xyzs     = (const float*)d_in[0];
    const float* features = (const float*)d_in[1];
    const float* w1  = (const float*)d_in[2];
    const float* b1  = (const float*)d_in[3];
    const float* g1  = (const float*)d_in[4];
    const float* be1 = (const float*)d_in[5];
    const float* w2  = (const float*)d_in[6];
    const float* b2  = (const float*)d_in[7];
    const float* dw1 = (const float*)d_in[8];
    const float* db1 = (const float*)d_in[9];
    const float* dg  = (const float*)d_in[10];
    const float* dbe = (const float*)d_in[11];
    const float* dw2 = (const float*)d_in[12];
    const float* db2 = (const float*)d_in[13];

    // workspace: h (B,N,COUT) then w (B,N)
    float* h_ws = (float*)d_ws;
    float* w_ws = h_ws + (size_t)BATCH * NPTS * COUT;

    // output: xyz (B,OUT_K,3) | feats (B,COUT,OUT_K) | indices (B,OUT_K) int32
    float* xyz_out   = (float*)d_out;
    float* feats_out = xyz_out + (size_t)BATCH * OUT_K * 3;
    int*   idx_out   = (int*)(feats_out + (size_t)BATCH * COUT * OUT_K);

    // 2048 row-tiles, 8 waves (one tile each) per 256-thread block.
    mlp_kernel<<<256, 256, 0, stream>>>(features, w1, b1, g1, be1, w2, b2,
                                        dw1, db1, dg, dbe, dw2, db2,
                                        h_ws, w_ws);
    topk_gather_kernel<<<BATCH, 1024, 0, stream>>>(w_ws, h_ws, xyzs,
                                                   xyz_out, feats_out, idx_out);
}